// ssgpr_rep_sm_51230369907346
// MI455X (gfx1250) — compile-verified
//
#include <hip/hip_runtime.h>
#include <hip/hip_bf16.h>
#include <math.h>

typedef __attribute__((ext_vector_type(2))) float v2f;
typedef __attribute__((ext_vector_type(8))) float v8f;

#define S_TOTAL 600
#define S_TILES 38             // ceil(600/16); last tile overlaps (base 584)
#define D_DIM   8
#define OUT_COLS 1200
#define ROW_TILES_PER_WAVE 4   // each wave covers 64 rows per col tile
#define ROWS_PER_BLOCK (8 * ROW_TILES_PER_WAVE * 16)   // 512

// ---------------------------------------------------------------------------
// Setup: build spectral points, per-column scale, per-column output indices.
//   spectral[s][d] = mu[q(s)][d] + std[q(s)][d] * eps[s][d]
//   scale[s]       = sqrt(weight[q]/count[q])
//   coscol[s]      = s + OFFSETS[q]      (cos block of component q)
//   sincol[s]      = s + OFFSETS[q+1]    (sin block of component q)
// ---------------------------------------------------------------------------
__global__ void ssgpr_setup_kernel(const float* __restrict__ weight,
                                   const float* __restrict__ mu,
                                   const float* __restrict__ stdv,
                                   const float* __restrict__ eps,
                                   float* __restrict__ spec,
                                   float* __restrict__ scale,
                                   int* __restrict__ coscol,
                                   int* __restrict__ sincol) {
  const int s = blockIdx.x * blockDim.x + threadIdx.x;
  if (s >= S_TOTAL) return;

  const int OFFS[7] = {0, 150, 270, 370, 460, 540, 600};

  int q = 0;
  #pragma unroll
  for (int i = 0; i < 5; ++i)
    if (s >= OFFS[i + 1]) q = i + 1;

  const int cnt = OFFS[q + 1] - OFFS[q];
  #pragma unroll
  for (int d = 0; d < D_DIM; ++d)
    spec[s * D_DIM + d] = mu[q * D_DIM + d] + stdv[q * D_DIM + d] * eps[s * D_DIM + d];

  scale[s]  = sqrtf(weight[q] / (float)cnt);
  coscol[s] = s + OFFS[q];
  sincol[s] = s + OFFS[q + 1];
}

// ---------------------------------------------------------------------------
// Main kernel: each wave32 owns one 16-column tile of spectral and loops over
// 4 row tiles of 16. Per row tile: two chained V_WMMA_F32_16X16X4_F32 build
// the 16x16 fp32 xdot tile (K=8), then hardware revolution-domain sin/cos
// (v_sin/v_cos compute sin(2*pi*x), exactly the reference argument), scale,
// and scatter into the ragged [cos_q | sin_q] layout.
//
// Edge handling via OVERLAPPING tiles (no guards, no per-lane 64-bit bases):
//  - last col tile base clamped to S_TOTAL-16: overlap columns recompute
//    bit-identical values and store identical data (benign duplicate writes)
//  - row tile base clamped to N-16 (wave-uniform s_min): same argument.
// All store indices are 32-bit offsets from the uniform `out` base, so the
// epilogue is add-only address math. EXEC stays all-1s (WMMA requirement).
//
// Layouts (CDNA5 ISA 7.12.2, f32):
//   A 16x4 : lane m=lane&15; VGPR0 holds K=kb, VGPR1 K=kb+1; kb = lane<16?0:2
//   B 4x16 : lane n=lane&15; VGPR0 holds K=kb, VGPR1 K=kb+1 (row striped)
//   C 16x16: VGPR v: lanes 0-15 -> M=v, lanes 16-31 -> M=v+8; N=lane&15
// ---------------------------------------------------------------------------
__global__ void __launch_bounds__(256)
ssgpr_wmma_kernel(const float* __restrict__ x,
                  const float* __restrict__ spec,
                  const float* __restrict__ scale,
                  const int* __restrict__ coscol,
                  const int* __restrict__ sincol,
                  float* __restrict__ out,
                  int N) {
  const int lane = threadIdx.x & 31;
  const int wave = threadIdx.x >> 5;          // 8 waves per block
  const int n    = lane & 15;                 // tile column (also A row m)
  const int kb   = (lane >> 4) << 1;          // 0 for lanes 0-15, 2 for 16-31
  const int mofs = (lane >> 4) << 3;          // lanes 16-31 own M = v+8

  // Column tile base, clamped so the last tile overlaps (all cols valid).
  int scol = blockIdx.x * 16;
  scol = scol < (S_TOTAL - 16) ? scol : (S_TOTAL - 16);

  const int wavebase = blockIdx.y * ROWS_PER_BLOCK +
                       wave * (ROW_TILES_PER_WAVE * 16);
  if (wavebase >= N) return;                  // wave-uniform

  // --- B operands + column metadata: invariant over the row-tile loop ------
  const int soff = (scol + n) * D_DIM;        // 32-bit offset
  v2f b0, b1;
  b0.x = spec[soff + kb];     b0.y = spec[soff + kb + 1];
  b1.x = spec[soff + kb + 4]; b1.y = spec[soff + kb + 5];

  const float sc   = scale[scol + n];
  const int   cc   = coscol[scol + n];
  const int   dcol = sincol[scol + n] - cc;   // sin offset within a row

  #pragma unroll
  for (int t = 0; t < ROW_TILES_PER_WAVE; ++t) {
    int rb = wavebase + t * 16;
    if (rb >= N) break;                       // wave-uniform
    rb = rb < (N - 16) ? rb : (N - 16);       // uniform clamp: overlap tile

    // --- A operands (all rows valid after clamp; 32-bit offsets) ----------
    const int xoff = (rb + n) * D_DIM;
    v2f a0, a1;
    a0.x = x[xoff + kb];     a0.y = x[xoff + kb + 1];
    a1.x = x[xoff + kb + 4]; a1.y = x[xoff + kb + 5];

    // --- D = x * spectral^T (K=8 via two chained K=4 WMMAs), fp32 ---------
    v8f c = {};
    c = __builtin_amdgcn_wmma_f32_16x16x4_f32(false, a0, false, b0,
                                              (short)0, c, false, false);
    c = __builtin_amdgcn_wmma_f32_16x16x4_f32(false, a1, false, b1,
                                              (short)0, c, false, false);

    // --- branchless epilogue: 32-bit add-only addressing -------------------
    int cidx = (rb + mofs) * OUT_COLS + cc;
    #pragma unroll
    for (int v = 0; v < 8; ++v) {
      const float d  = c[v];                           // argument / (2*pi)
      const float cv = __builtin_amdgcn_cosf(d);       // cos(2*pi*d)
      const float sv = __builtin_amdgcn_sinf(d);       // sin(2*pi*d)
      out[cidx]        = sc * cv;
      out[cidx + dcol] = sc * sv;
      cidx += OUT_COLS;
    }
  }
}

// ---------------------------------------------------------------------------
// Launch
// ---------------------------------------------------------------------------
extern "C" void kernel_launch(void* const* d_in, const int* in_sizes, int n_in,
                              void* d_out, int out_size, void* d_ws, size_t ws_size,
                              hipStream_t stream) {
  const float* x      = (const float*)d_in[0];   // [N, 8]
  const float* weight = (const float*)d_in[1];   // [6]
  const float* mu     = (const float*)d_in[2];   // [6, 8]
  const float* stdv   = (const float*)d_in[3];   // [6, 8]
  const float* eps    = (const float*)d_in[4];   // [600, 8]
  float* out = (float*)d_out;                    // [N, 1200]

  const int N = in_sizes[0] / D_DIM;

  // Workspace layout (floats/ints in d_ws)
  float* spec   = (float*)d_ws;                       // 600*8 floats
  float* scale  = spec + (size_t)S_TOTAL * D_DIM;     // 600 floats
  int*   coscol = (int*)(scale + S_TOTAL);            // 600 ints
  int*   sincol = coscol + S_TOTAL;                   // 600 ints

  ssgpr_setup_kernel<<<(S_TOTAL + 255) / 256, 256, 0, stream>>>(
      weight, mu, stdv, eps, spec, scale, coscol, sincol);

  dim3 grid(S_TILES, (N + ROWS_PER_BLOCK - 1) / ROWS_PER_BLOCK);
  dim3 block(256);                          // 8 wave32
  ssgpr_wmma_kernel<<<grid, block, 0, stream>>>(
      x, spec, scale, coscol, sincol, out, N);
}